// TopologicalNeuralArchitectures_39737037422992
// MI455X (gfx1250) — compile-verified
//
#include <hip/hip_runtime.h>
#include <hip/hip_bf16.h>
#include <cstddef>
#include <cstdint>

typedef __attribute__((ext_vector_type(2))) float v2f;
typedef __attribute__((ext_vector_type(8))) float v8f;
typedef __attribute__((ext_vector_type(4))) unsigned int v4u;
typedef __attribute__((ext_vector_type(8))) int v8i;
typedef __attribute__((ext_vector_type(4))) int v4i;

#if defined(__HIP_DEVICE_COMPILE__) && !__has_builtin(__builtin_amdgcn_wmma_f32_16x16x4_f32)
#error "wmma f32 16x16x4 builtin not available (device pass)"
#endif
#if defined(__HIP_DEVICE_COMPILE__) && __has_builtin(__builtin_amdgcn_tensor_load_to_lds)
#define HAVE_TDM 1
#endif

#define Ldim 3
#define Bdim 64
#define Ndim 64
#define Ddim 64
#define Edim 129

// LDS row strides (padded for conflict-free WMMA fragment reads).
#define LDA 132   // update kernel A panel: 64 rows x 128 K
#define LDB 72    // W panel: 128 rows x 64 N  (64 DW + 8 DW pad)
#define LDS_SJ 68 // S panels: 64 rows x 64 K  (64 DW + 4 DW pad)

// ---- WMMA fragment loads (V_WMMA_F32_16X16X4_F32 layout) ------------------
// lanes 0-15: VGPR0=A[m][k0], VGPR1=A[m][k0+1]; lanes 16-31: K=k0+2,k0+3
__device__ __forceinline__ v2f ldA_s(const float* base, int stride, int mbase, int k0, int lane) {
  int m = mbase + (lane & 15);
  int k = k0 + ((lane >> 4) << 1);
  const float* p = base + m * stride + k;
  v2f r = {p[0], p[1]};
  return r;
}
__device__ __forceinline__ v2f ldB(const float* sW, int k0, int nbase, int lane) {
  int n = nbase + (lane & 15);
  int k = k0 + ((lane >> 4) << 1);
  v2f r = {sW[k * LDB + n], sW[(k + 1) * LDB + n]};
  return r;
}
__device__ __forceinline__ v8f wmma_f32(v2f a, v2f b, v8f c) {
#if defined(__HIP_DEVICE_COMPILE__)
  return __builtin_amdgcn_wmma_f32_16x16x4_f32(false, a, false, b, (short)0, c, false, false);
#else
  return c;
#endif
}

// ---- Tensor Data Mover: 2-D tile (f32) global -> LDS with row padding -----
// D# packing per CDNA5 ISA. 6-arg builtin form
// (clang-23 toolchain): (v4u g0, v8i g1, v4i g2, v4i g3, v8i extra, i32 cpol).
//   tile_w   : elements per row (dim0), tile_h: rows (dim1)
//   g_stride : global row stride in elements
//   pad_i/pad_a : LDS pad codes (interval = 2^(code+1) DW, amount = code+1 DW)
__device__ __forceinline__ void tdm_load_2d(unsigned lds_off, const float* gptr,
                                            unsigned tile_w, unsigned tile_h,
                                            unsigned g_stride,
                                            unsigned pad_i, unsigned pad_a) {
#if defined(HAVE_TDM)
  unsigned long long ga = (unsigned long long)(uintptr_t)gptr;
  v4u g0;
  g0[0] = 1u;                                        // count=1, user descriptor
  g0[1] = lds_off;                                   // lds_addr (bytes)
  g0[2] = (unsigned)(ga & 0xffffffffu);              // global_addr[31:0]
  g0[3] = (unsigned)((ga >> 32) & 0x01ffffffu)       // global_addr[56:32]
        | (2u << 30);                                // type = 2 ("image")
  v8i g1;
  g1[0] = (int)((2u << 16)                           // data_size = 4 bytes
        | (1u << 20)                                 // pad_enable
        | (pad_i << 22) | (pad_a << 25));
  g1[1] = (int)(tile_w << 16);                       // tensor_dim0[15:0]
  g1[2] = (int)(tile_h << 16);                       // tensor_dim0[31:16]=0 | tensor_dim1[15:0]
  g1[3] = (int)(tile_w << 16);                       // tensor_dim1[31:16]=0 | tile_dim0
  g1[4] = (int)tile_h;                               // tile_dim1 | tile_dim2=0
  g1[5] = (int)g_stride;                             // tensor_dim0_stride[31:0]
  g1[6] = 0;                                         // stride[47:32]=0 | dim1_stride lo
  g1[7] = 0;
  v4i z4 = {0, 0, 0, 0};
  v8i z8 = {0, 0, 0, 0, 0, 0, 0, 0};
  __builtin_amdgcn_tensor_load_to_lds(g0, g1, z4, z4, z8, 0);
#else
  (void)lds_off; (void)gptr; (void)tile_w; (void)tile_h;
  (void)g_stride; (void)pad_i; (void)pad_a;
#endif
}
__device__ __forceinline__ void wait_tensorcnt_le2() {
#if defined(HAVE_TDM)
  __builtin_amdgcn_s_wait_tensorcnt((short)2);
#endif
}
__device__ __forceinline__ void wait_tensorcnt_0() {
#if defined(HAVE_TDM)
  __builtin_amdgcn_s_wait_tensorcnt((short)0);
#endif
}
__device__ __forceinline__ unsigned lds_offset_of(const void* p) {
  return (unsigned)(uintptr_t)p;   // generic ptr low 32 bits = LDS offset
}

// ---------------------------------------------------------------------------
// Prep: adjacency -> neighbor lists, counts, readout coefficients c[i]
// ---------------------------------------------------------------------------
__global__ void prep_kernel(const float* __restrict__ ew, int* __restrict__ nbr_cnt,
                            int* __restrict__ nbr_idx, float* __restrict__ cvec) {
  __shared__ float rowsum[Ndim];
  __shared__ float stot;
  const int i = threadIdx.x;  // 64 threads
  int cnt = 0;
  float rs = 0.f;
  for (int j = 0; j < Ndim; ++j) {
    float w = ew[i * Ndim + j];
    if (w > 0.f) { nbr_idx[i * Ndim + cnt] = j; ++cnt; }
    rs += w;
  }
  nbr_cnt[i] = cnt;
  rowsum[i] = rs;
  __syncthreads();
  if (i == 0) {
    float t = 0.f;
    for (int k = 0; k < Ndim; ++k) t += rowsum[k];
    stot = t;
  }
  __syncthreads();
  cvec[i] = rs / (stot + 1e-8f);
}

// ---------------------------------------------------------------------------
// Message + aggregation, edges only. One workgroup per node i.
// TDM double-buffers the per-edge W panel (128x64) and S_j panel (64x64);
// wave 0 prefetches edge e+1 while all 8 waves run WMMA on edge e.
// ---------------------------------------------------------------------------
__global__ __launch_bounds__(256)
void msg_agg_kernel(const float* __restrict__ s_in, const float* __restrict__ Wm,
                    const float* __restrict__ bmsg, const float* __restrict__ ew,
                    const int* __restrict__ nbr_cnt, const int* __restrict__ nbr_idx,
                    float* __restrict__ agg) {
  extern __shared__ float smem[];
  float* sSi = smem;                         // 64 x LDS_SJ
  float* sSj = sSi + Bdim * LDS_SJ;          // 2 x 64 x LDS_SJ
  float* sW  = sSj + 2 * Bdim * LDS_SJ;      // 2 x 128 x LDB
  float* sC  = sW  + 2 * 128 * LDB;          // 64 column constants
  const int tid = threadIdx.x;
  const int i = blockIdx.x;
  const int wave = tid >> 5, lane = tid & 31;

  // Stage S_i once (vector loads; outside the edge loop)
  for (int r = 0; r < 4; ++r) {
    int q = tid + 256 * r;                   // 1024 float4
    int b = q >> 4, d = (q & 15) << 2;
    float4 v = *(const float4*)(s_in + ((size_t)b * Ndim + i) * Ddim + d);
    float* dst = sSi + b * LDS_SJ + d;
    dst[0] = v.x; dst[1] = v.y; dst[2] = v.z; dst[3] = v.w;
  }

  float cacc = 0.f;
  v8f c0 = {}, c1 = {};
  const int dn  = (wave & 3) * 16;
  const int bm0 = ((wave >> 2) * 2) * 16;
  const int bm1 = bm0 + 16;
  const int cnt = nbr_cnt[i];

  // Prologue: prefetch edge 0 into buffer 0 (wave 0 only; EXEC-independent DMA)
  if (cnt > 0 && wave == 0) {
    int j0 = nbr_idx[i * Ndim];
    tdm_load_2d(lds_offset_of(sW), Wm + ((size_t)i * Ndim + j0) * Edim * Ddim,
                64, 128, 64, /*pad 64DW*/5, /*+8DW*/7);
    tdm_load_2d(lds_offset_of(sSj), s_in + (size_t)j0 * Ddim,
                64, 64, Ndim * Ddim, /*pad 64DW*/5, /*+4DW*/3);
  }

  for (int e = 0; e < cnt; ++e) {
    const int cur = e & 1;
    __syncthreads();                         // buffer cur^1 free (edge e-1 compute done)
    if (wave == 0 && e + 1 < cnt) {          // prefetch next edge into cur^1
      int j1 = nbr_idx[i * Ndim + e + 1];
      tdm_load_2d(lds_offset_of(sW + (cur ^ 1) * 128 * LDB),
                  Wm + ((size_t)i * Ndim + j1) * Edim * Ddim, 64, 128, 64, 5, 7);
      tdm_load_2d(lds_offset_of(sSj + (cur ^ 1) * Bdim * LDS_SJ),
                  s_in + (size_t)j1 * Ddim, 64, 64, Ndim * Ddim, 5, 3);
    }
    if (tid < Ddim) {                        // ew*w3 + bias (global reads, no LDS dep)
      int j = nbr_idx[i * Ndim + e];
      const float* wb = Wm + ((size_t)i * Ndim + j) * Edim * Ddim;
      cacc += ew[i * Ndim + j] * wb[(size_t)128 * Ddim + tid]
            + bmsg[((size_t)i * Ndim + j) * Ddim + tid];
    }
    if (wave == 0) {                         // edge-e transfers complete
      if (e + 1 < cnt) wait_tensorcnt_le2(); else wait_tensorcnt_0();
    }
    __syncthreads();                         // publish edge-e panels to all waves

    const float* sWc = sW + cur * 128 * LDB;
    const float* sJc = sSj + cur * Bdim * LDS_SJ;
    #pragma unroll 8
    for (int k0 = 0; k0 < 64; k0 += 4) {     // K = feat_i half (from S_i panel)
      v2f a0 = ldA_s(sSi, LDS_SJ, bm0, k0, lane);
      v2f a1 = ldA_s(sSi, LDS_SJ, bm1, k0, lane);
      v2f bf = ldB(sWc, k0, dn, lane);
      c0 = wmma_f32(a0, bf, c0);
      c1 = wmma_f32(a1, bf, c1);
    }
    #pragma unroll 8
    for (int k0 = 64; k0 < 128; k0 += 4) {   // K = feat_j half (from S_j panel)
      v2f a0 = ldA_s(sJc, LDS_SJ, bm0, k0 - 64, lane);
      v2f a1 = ldA_s(sJc, LDS_SJ, bm1, k0 - 64, lane);
      v2f bf = ldB(sWc, k0, dn, lane);
      c0 = wmma_f32(a0, bf, c0);
      c1 = wmma_f32(a1, bf, c1);
    }
  }

  if (tid < Ddim) sC[tid] = cacc;
  __syncthreads();

  const int dcol = dn + (lane & 15);
  const float cst = (cnt > 0) ? sC[dcol] : 0.f;
  const int half = lane >> 4;
  #pragma unroll
  for (int r = 0; r < 8; ++r) {
    int br = bm0 + r + 8 * half;
    agg[((size_t)br * Ndim + i) * Ddim + dcol] = c0[r] + cst;
    br = bm1 + r + 8 * half;
    agg[((size_t)br * Ndim + i) * Ddim + dcol] = c1[r] + cst;
  }
}

// ---------------------------------------------------------------------------
// Update: s = 0.9*([s|agg] @ W_upd + b_upd) + 0.1*s   (one workgroup per node)
// ---------------------------------------------------------------------------
__global__ __launch_bounds__(256)
void update_kernel(const float* __restrict__ s_in, const float* __restrict__ agg,
                   const float* __restrict__ Wu, const float* __restrict__ bu,
                   float* __restrict__ s_out) {
  extern __shared__ float smem[];
  float* sA = smem;                  // 64 x LDA
  float* sW = sA + Bdim * LDA;       // 128 x LDB
  const int tid = threadIdx.x;
  const int n = blockIdx.x;
  const int wave = tid >> 5, lane = tid & 31;

  for (int r = 0; r < 4; ++r) {
    int q = tid + 256 * r;
    int b = q >> 4, d = (q & 15) << 2;
    float4 v = *(const float4*)(s_in + ((size_t)b * Ndim + n) * Ddim + d);
    float* dst = sA + b * LDA + d;
    dst[0] = v.x; dst[1] = v.y; dst[2] = v.z; dst[3] = v.w;
    float4 va = *(const float4*)(agg + ((size_t)b * Ndim + n) * Ddim + d);
    float* dsta = sA + b * LDA + 64 + d;
    dsta[0] = va.x; dsta[1] = va.y; dsta[2] = va.z; dsta[3] = va.w;
  }
  for (int r = 0; r < 8; ++r) {
    int q = tid + 256 * r;
    int e = q >> 4, d = (q & 15) << 2;
    float4 v = *(const float4*)(Wu + (size_t)e * Ddim + d);
    float* dst = sW + e * LDB + d;
    dst[0] = v.x; dst[1] = v.y; dst[2] = v.z; dst[3] = v.w;
  }
  __syncthreads();

  v8f c0 = {}, c1 = {};
  const int dn  = (wave & 3) * 16;
  const int bm0 = ((wave >> 2) * 2) * 16, bm1 = bm0 + 16;
  #pragma unroll 8
  for (int k0 = 0; k0 < 128; k0 += 4) {
    v2f a0 = ldA_s(sA, LDA, bm0, k0, lane);
    v2f a1 = ldA_s(sA, LDA, bm1, k0, lane);
    v2f bf = ldB(sW, k0, dn, lane);
    c0 = wmma_f32(a0, bf, c0);
    c1 = wmma_f32(a1, bf, c1);
  }
  const int dcol = dn + (lane & 15);
  const float bud = bu[dcol];
  const int half = lane >> 4;
  #pragma unroll
  for (int r = 0; r < 8; ++r) {
    int br = bm0 + r + 8 * half;
    float sv = sA[br * LDA + dcol];
    s_out[((size_t)br * Ndim + n) * Ddim + dcol] = 0.9f * (c0[r] + bud) + 0.1f * sv;
    br = bm1 + r + 8 * half;
    sv = sA[br * LDA + dcol];
    s_out[((size_t)br * Ndim + n) * Ddim + dcol] = 0.9f * (c1[r] + bud) + 0.1f * sv;
  }
}

// ---------------------------------------------------------------------------
// Sequential smoothing scan: batch-parallel, 8 batches per WG in LDS (128 KB),
// 64 serial node steps with workgroup barriers.
// ---------------------------------------------------------------------------
__global__ __launch_bounds__(256)
void smooth_kernel(float* __restrict__ s, const int* __restrict__ nbr_cnt,
                   const int* __restrict__ nbr_idx) {
  extern __shared__ float st[];      // [8][64][64]
  const int tid = threadIdx.x;
  const int b0 = blockIdx.x * 8;
  for (int r = 0; r < 32; ++r) {
    int q = tid + 256 * r;           // 8192 float4
    int bl = q >> 10, rem = q & 1023;
    int n = rem >> 4, d = (rem & 15) << 2;
    float4 v = *(const float4*)(s + (((size_t)(b0 + bl)) * Ndim + n) * Ddim + d);
    float* dst = st + ((bl * Ndim + n) * Ddim + d);
    dst[0] = v.x; dst[1] = v.y; dst[2] = v.z; dst[3] = v.w;
  }
  __syncthreads();
  for (int i = 0; i < Ndim; ++i) {
    const int cnt = nbr_cnt[i];
    if (cnt > 0) {
      float nv0, nv1;
      {
        int q = tid;               int bl = q >> 6, d = q & 63;
        float sum = 0.f;
        for (int e = 0; e < cnt; ++e) sum += st[(bl * Ndim + nbr_idx[i * Ndim + e]) * Ddim + d];
        nv0 = 0.95f * st[(bl * Ndim + i) * Ddim + d] + 0.05f * (sum / (float)cnt);
      }
      {
        int q = tid + 256;         int bl = q >> 6, d = q & 63;
        float sum = 0.f;
        for (int e = 0; e < cnt; ++e) sum += st[(bl * Ndim + nbr_idx[i * Ndim + e]) * Ddim + d];
        nv1 = 0.95f * st[(bl * Ndim + i) * Ddim + d] + 0.05f * (sum / (float)cnt);
      }
      { int q = tid;       st[((q >> 6) * Ndim + i) * Ddim + (q & 63)] = nv0; }
      { int q = tid + 256; st[((q >> 6) * Ndim + i) * Ddim + (q & 63)] = nv1; }
    }
    __syncthreads();
  }
  for (int r = 0; r < 32; ++r) {
    int q = tid + 256 * r;
    int bl = q >> 10, rem = q & 1023;
    int n = rem >> 4, d = (rem & 15) << 2;
    float* src = st + ((bl * Ndim + n) * Ddim + d);
    *(float4*)(s + (((size_t)(b0 + bl)) * Ndim + n) * Ddim + d) =
        make_float4(src[0], src[1], src[2], src[3]);
  }
}

// ---------------------------------------------------------------------------
// Readout: g[b,:] = sum_i c[i]*s[b,i,:];  out = g @ W_out + b_out
// ---------------------------------------------------------------------------
__global__ __launch_bounds__(256)
void final_kernel(const float* __restrict__ s, const float* __restrict__ cvec,
                  const float* __restrict__ Wo, const float* __restrict__ bo,
                  float* __restrict__ out) {
  __shared__ float sG[Bdim * Ddim];
  __shared__ float sWo[Ddim * Ddim];
  const int tid = threadIdx.x;
  for (int r = 0; r < 16; ++r) {
    int q = tid + 256 * r;
    int b = q >> 6, d = q & 63;
    float acc = 0.f;
    for (int i = 0; i < Ndim; ++i)
      acc += s[(((size_t)b) * Ndim + i) * Ddim + d] * cvec[i];
    sG[q] = acc;
    sWo[q] = Wo[q];
  }
  __syncthreads();
  for (int r = 0; r < 16; ++r) {
    int q = tid + 256 * r;
    int b = q >> 6, d2 = q & 63;
    float acc = bo[d2];
    for (int d = 0; d < Ddim; ++d) acc += sG[b * Ddim + d] * sWo[d * Ddim + d2];
    out[q] = acc;
  }
}

extern "C" void kernel_launch(void* const* d_in, const int* in_sizes, int n_in,
                              void* d_out, int out_size, void* d_ws, size_t ws_size,
                              hipStream_t stream) {
  (void)in_sizes; (void)n_in; (void)out_size; (void)ws_size;
  const float* x  = (const float*)d_in[0];
  const float* ew = (const float*)d_in[1];
  const float* Wm = (const float*)d_in[2];
  const float* bm = (const float*)d_in[3];
  const float* Wu = (const float*)d_in[4];
  const float* bu = (const float*)d_in[5];
  const float* Wo = (const float*)d_in[6];
  const float* bo = (const float*)d_in[7];
  float* out = (float*)d_out;

  char* ws = (char*)d_ws;
  float* s_ws   = (float*)ws; ws += (size_t)Bdim * Ndim * Ddim * sizeof(float);
  float* agg    = (float*)ws; ws += (size_t)Bdim * Ndim * Ddim * sizeof(float);
  int*   nbridx = (int*)ws;   ws += (size_t)Ndim * Ndim * sizeof(int);
  int*   nbrcnt = (int*)ws;   ws += (size_t)Ndim * sizeof(int);
  float* cvec   = (float*)ws;

  prep_kernel<<<1, 64, 0, stream>>>(ew, nbrcnt, nbridx, cvec);

  const size_t shA = (size_t)(3 * Bdim * LDS_SJ + 2 * 128 * LDB + Ddim) * sizeof(float);
  const size_t shU = (size_t)(Bdim * LDA + 128 * LDB) * sizeof(float);
  const size_t shS = (size_t)8 * Ndim * Ddim * sizeof(float);

  for (int l = 0; l < Ldim; ++l) {
    const float* s_in = (l == 0) ? x : s_ws;
    const float* Wml = Wm + (size_t)l * Ndim * Ndim * Edim * Ddim;
    const float* bml = bm + (size_t)l * Ndim * Ndim * Ddim;
    const float* Wul = Wu + (size_t)l * 2 * Ddim * Ddim;
    const float* bul = bu + (size_t)l * Ddim;
    msg_agg_kernel<<<Ndim, 256, shA, stream>>>(s_in, Wml, bml, ew, nbrcnt, nbridx, agg);
    update_kernel<<<Ndim, 256, shU, stream>>>(s_in, agg, Wul, bul, s_ws);
    smooth_kernel<<<Bdim / 8, 256, shS, stream>>>(s_ws, nbrcnt, nbridx);
  }
  final_kernel<<<1, 256, 0, stream>>>(s_ws, cvec, Wo, bo, out);
}